// GAT_37014028157431
// MI455X (gfx1250) — compile-verified
//
#include <hip/hip_runtime.h>
#include <hip/hip_bf16.h>
#include <cmath>

typedef __attribute__((ext_vector_type(2))) float v2f;
typedef __attribute__((ext_vector_type(8))) float v8f;

#define NEG_SLOPE 0.2f

__device__ __forceinline__ float lrelu(float x) { return x > 0.f ? x : NEG_SLOPE * x; }

// sign-trick float atomic max (works with -inf init, no CAS loop)
__device__ __forceinline__ void atomicMaxF(float* addr, float v) {
  if (v >= 0.f) atomicMax((int*)addr, __float_as_int(v));
  else          atomicMin((unsigned int*)addr, __float_as_uint(v));
}

// decode edge e of the (E real edges + N self loops) list
__device__ __forceinline__ void edge_sd(const long long* __restrict__ src,
                                        const long long* __restrict__ dst,
                                        int e, int E, int& s, int& d) {
  if (e < E) { s = (int)src[e]; d = (int)dst[e]; }
  else       { s = e - E; d = e - E; }
}

// ---------------------------------------------------------------------------
// GEMM  C[M x Ncol] = A[M x K] @ W[K x Ncol]   (f32, WMMA 16x16x4)
// block = 128 threads (4 waves), block tile 16 rows x 64 cols, K chunked by 128.
// Both A (row-major) and B (transposed: ldsB[col][k]) are staged in LDS with a
// 132-float row pitch (lane stride = 4 banks -> conflict-free b64 fragment
// loads). Inner loop = 2 LDS b64 loads + 1 WMMA, no address arithmetic.
// ---------------------------------------------------------------------------
#define KC 128
__global__ __launch_bounds__(128) void gemm_wmma(const float* __restrict__ A,
                                                 const float* __restrict__ W,
                                                 float* __restrict__ C,
                                                 int M, int K, int Ncol) {
  __shared__ float ldsA[16 * 132];
  __shared__ float ldsB[64 * 132];
  const int row0    = blockIdx.x * 16;
  const int wave    = threadIdx.x >> 5;
  const int lane    = threadIdx.x & 31;
  const int colbase = blockIdx.y * 64;
  const int rsel    = lane & 15;          // A row / B col selector
  const int kk      = (lane >> 4) * 2;    // K sub-offset per lane half

  v8f c = {};
  for (int kc = 0; kc < K; kc += KC) {
    __syncthreads();   // protect LDS from previous chunk's readers

    // stage A tile: 16 x KC (float4, coalesced)
    for (int i = threadIdx.x; i < 16 * (KC / 4); i += 128) {
      int r = i / (KC / 4), cs = i % (KC / 4);
      int ra = row0 + r; if (ra >= M) ra = M - 1;
      float4 v = ((const float4*)(A + (size_t)ra * K + kc))[cs];
      *(float4*)&ldsA[r * 132 + cs * 4] = v;
    }
    // stage B tile transposed: ldsB[c][k] = W[kc+k][colbase+c]
    for (int i = threadIdx.x; i < KC * 16; i += 128) {
      int k = i >> 4, c4 = (i & 15) << 2;
      float4 v = *(const float4*)(W + (size_t)(kc + k) * Ncol + colbase + c4);
      ldsB[(c4 + 0) * 132 + k] = v.x;
      ldsB[(c4 + 1) * 132 + k] = v.y;
      ldsB[(c4 + 2) * 132 + k] = v.z;
      ldsB[(c4 + 3) * 132 + k] = v.w;
    }
    __syncthreads();

    const float* ap = &ldsA[rsel * 132 + kk];
    const float* bp = &ldsB[(wave * 16 + rsel) * 132 + kk];
#pragma unroll
    for (int k = 0; k < KC; k += 4) {
      v2f a, b;
      a.x = ap[k]; a.y = ap[k + 1];
      b.x = bp[k]; b.y = bp[k + 1];
      c = __builtin_amdgcn_wmma_f32_16x16x4_f32(false, a, false, b,
                                                (short)0, c, false, false);
    }
  }

  const int crow = (lane >> 4) * 8;
#pragma unroll
  for (int r = 0; r < 8; ++r) {
    int rr = row0 + crow + r;
    if (rr < M) C[(size_t)rr * Ncol + colbase + wave * 16 + rsel] = c[r];
  }
}

// ---------------------------------------------------------------------------
// attention logits: alpha_s[n,h] = dot(h[n,h,:], a_src[h]);  same for dst
// ---------------------------------------------------------------------------
__global__ __launch_bounds__(256) void alpha_kernel(const float* __restrict__ h,
                                                    const float* __restrict__ a_src,
                                                    const float* __restrict__ a_dst,
                                                    float* __restrict__ as,
                                                    float* __restrict__ ad,
                                                    int n, int heads, int ch) {
  int idx = blockIdx.x * blockDim.x + threadIdx.x;
  if (idx >= n * heads) return;
  int node = idx / heads, hd = idx % heads;
  const float* hp = h + (size_t)node * heads * ch + hd * ch;
  const float* sp = a_src + hd * ch;
  const float* dp = a_dst + hd * ch;
  float accs = 0.f, accd = 0.f;
  for (int c = 0; c < ch; ++c) { float v = hp[c]; accs += v * sp[c]; accd += v * dp[c]; }
  as[idx] = accs;
  ad[idx] = accd;
}

__global__ void fill_f32(float* p, float v, int n) {
  int i = blockIdx.x * blockDim.x + threadIdx.x;
  if (i < n) p[i] = v;
}

__global__ void fill_bias(float* p, const float* __restrict__ b, int n, int ch) {
  int i = blockIdx.x * blockDim.x + threadIdx.x;
  if (i < n) p[i] = b[i % ch];
}

__global__ void relu_inplace(float* p, int n) {
  int i = blockIdx.x * blockDim.x + threadIdx.x;
  if (i < n) { float v = p[i]; p[i] = v > 0.f ? v : 0.f; }
}

// segment max over destination nodes
__global__ __launch_bounds__(256) void edge_max_k(const long long* __restrict__ src,
                                                  const long long* __restrict__ dst,
                                                  const float* __restrict__ as,
                                                  const float* __restrict__ ad,
                                                  float* __restrict__ m,
                                                  int E, int Et, int heads) {
  int idx = blockIdx.x * blockDim.x + threadIdx.x;
  if (idx >= Et * heads) return;
  int e = idx / heads, h = idx % heads;
  int s, d; edge_sd(src, dst, e, E, s, d);
  float v = lrelu(as[s * heads + h] + ad[d * heads + h]);
  atomicMaxF(&m[d * heads + h], v);
}

// segment sum of exp(e - m[dst])
__global__ __launch_bounds__(256) void edge_denom_k(const long long* __restrict__ src,
                                                    const long long* __restrict__ dst,
                                                    const float* __restrict__ as,
                                                    const float* __restrict__ ad,
                                                    const float* __restrict__ m,
                                                    float* __restrict__ dn,
                                                    int E, int Et, int heads) {
  int idx = blockIdx.x * blockDim.x + threadIdx.x;
  if (idx >= Et * heads) return;
  int e = idx / heads, h = idx % heads;
  int s, d; edge_sd(src, dst, e, E, s, d);
  float v = lrelu(as[s * heads + h] + ad[d * heads + h]);
  atomicAdd(&dn[d * heads + h], expf(v - m[d * heads + h]));
}

// weighted scatter-add: one block per edge, one thread per output channel
__global__ __launch_bounds__(256) void aggregate_k(const long long* __restrict__ src,
                                                   const long long* __restrict__ dst,
                                                   const float* __restrict__ as,
                                                   const float* __restrict__ ad,
                                                   const float* __restrict__ m,
                                                   const float* __restrict__ dn,
                                                   const float* __restrict__ hfeat,
                                                   float* __restrict__ out,
                                                   int E, int heads, int ch) {
  int e = blockIdx.x;
  int tid = threadIdx.x;            // blockDim == heads*ch
  int h = tid / ch;
  int s, d; edge_sd(src, dst, e, E, s, d);
  int hc = heads * ch;
  float v = lrelu(as[s * heads + h] + ad[d * heads + h]);
  float w = expf(v - m[d * heads + h]) / (dn[d * heads + h] + 1e-16f);
  atomicAdd(&out[(size_t)d * hc + tid], w * hfeat[(size_t)s * hc + tid]);
}

// regression head: one wave (32 lanes) per edge, dot([z_s, z_d], We) + be
__global__ __launch_bounds__(256) void edge_head_k(const long long* __restrict__ src,
                                                   const long long* __restrict__ dst,
                                                   const float* __restrict__ z,
                                                   const float* __restrict__ We,
                                                   const float* __restrict__ be,
                                                   float* __restrict__ out, int E) {
  int wv = blockIdx.x * (blockDim.x >> 5) + (threadIdx.x >> 5);
  if (wv >= E) return;
  int lane = threadIdx.x & 31;
  int s = (int)src[wv], d = (int)dst[wv];
  float acc = 0.f;
#pragma unroll
  for (int i = lane; i < 128; i += 32) {
    acc += z[(size_t)s * 128 + i] * We[i];
    acc += z[(size_t)d * 128 + i] * We[128 + i];
  }
#pragma unroll
  for (int off = 16; off > 0; off >>= 1) acc += __shfl_down(acc, off, 32);
  if (lane == 0) out[wv] = acc + be[0];
}

// ---------------------------------------------------------------------------
extern "C" void kernel_launch(void* const* d_in, const int* in_sizes, int n_in,
                              void* d_out, int out_size, void* d_ws, size_t ws_size,
                              hipStream_t stream) {
  const float*     x    = (const float*)d_in[0];
  const long long* ei   = (const long long*)d_in[1];
  const float*     W1   = (const float*)d_in[2];
  const float*     a_s1 = (const float*)d_in[3];
  const float*     a_d1 = (const float*)d_in[4];
  const float*     b1   = (const float*)d_in[5];
  const float*     W2   = (const float*)d_in[6];
  const float*     a_s2 = (const float*)d_in[7];
  const float*     a_d2 = (const float*)d_in[8];
  const float*     b2   = (const float*)d_in[9];
  const float*     We   = (const float*)d_in[10];
  const float*     be   = (const float*)d_in[11];
  float* out = (float*)d_out;

  const int N  = in_sizes[0] / 256;
  const int E  = in_sizes[1] / 2;
  const int Et = E + N;
  const long long* src = ei;       // edge_index[0]
  const long long* dst = ei + E;   // edge_index[1]

  float* ws  = (float*)d_ws;
  float* h1  = ws;                        // N*256
  float* o1  = h1  + (size_t)N * 256;     // N*256  (also reused as o2)
  float* as1 = o1  + (size_t)N * 256;     // N*8
  float* ad1 = as1 + (size_t)N * 8;
  float* m1  = ad1 + (size_t)N * 8;
  float* dn1 = m1  + (size_t)N * 8;
  float* as2 = dn1 + (size_t)N * 8;       // N
  float* ad2 = as2 + N;
  float* m2  = ad2 + N;
  float* dn2 = m2  + N;
  float* h2  = h1;                        // reuse: h1 dead after aggregate1
  float* o2  = o1;                        // reuse: o1 dead after gemm2

  // ---------- layer 1 (8 heads x 32) ----------
  gemm_wmma<<<dim3((N + 15) / 16, 4), 128, 0, stream>>>(x, W1, h1, N, 256, 256);
  alpha_kernel<<<(N * 8 + 255) / 256, 256, 0, stream>>>(h1, a_s1, a_d1, as1, ad1, N, 8, 32);
  fill_f32<<<(N * 8 + 255) / 256, 256, 0, stream>>>(m1, -INFINITY, N * 8);
  fill_f32<<<(N * 8 + 255) / 256, 256, 0, stream>>>(dn1, 0.f, N * 8);
  fill_bias<<<(N * 256 + 255) / 256, 256, 0, stream>>>(o1, b1, N * 256, 256);
  edge_max_k<<<(Et * 8 + 255) / 256, 256, 0, stream>>>(src, dst, as1, ad1, m1, E, Et, 8);
  edge_denom_k<<<(Et * 8 + 255) / 256, 256, 0, stream>>>(src, dst, as1, ad1, m1, dn1, E, Et, 8);
  aggregate_k<<<Et, 256, 0, stream>>>(src, dst, as1, ad1, m1, dn1, h1, o1, E, 8, 32);
  relu_inplace<<<(N * 256 + 255) / 256, 256, 0, stream>>>(o1, N * 256);

  // ---------- layer 2 (1 head x 128) ----------
  gemm_wmma<<<dim3((N + 15) / 16, 2), 128, 0, stream>>>(o1, W2, h2, N, 256, 128);
  alpha_kernel<<<(N + 255) / 256, 256, 0, stream>>>(h2, a_s2, a_d2, as2, ad2, N, 1, 128);
  fill_f32<<<(N + 255) / 256, 256, 0, stream>>>(m2, -INFINITY, N);
  fill_f32<<<(N + 255) / 256, 256, 0, stream>>>(dn2, 0.f, N);
  fill_bias<<<(N * 128 + 255) / 256, 256, 0, stream>>>(o2, b2, N * 128, 128);
  edge_max_k<<<(Et + 255) / 256, 256, 0, stream>>>(src, dst, as2, ad2, m2, E, Et, 1);
  edge_denom_k<<<(Et + 255) / 256, 256, 0, stream>>>(src, dst, as2, ad2, m2, dn2, E, Et, 1);
  aggregate_k<<<Et, 128, 0, stream>>>(src, dst, as2, ad2, m2, dn2, h2, o2, E, 1, 128);

  // ---------- edge regression head ----------
  edge_head_k<<<(E + 7) / 8, 256, 0, stream>>>(src, dst, o2, We, be, out, E);
}